// ReIDLoss_76802605187579
// MI455X (gfx1250) — compile-verified
//
#include <hip/hip_runtime.h>
#include <hip/hip_bf16.h>

typedef __attribute__((ext_vector_type(2))) float v2f;
typedef __attribute__((ext_vector_type(8))) float v8f;

#define REID_EPS 1e-6f

// Select element i (0..7) from an 8-wide float vector with a runtime index.
__device__ __forceinline__ float vsel8(v8f v, int i) {
    float r = v[0];
    r = (i == 1) ? v[1] : r;
    r = (i == 2) ? v[2] : r;
    r = (i == 3) ? v[3] : r;
    r = (i == 4) ? v[4] : r;
    r = (i == 5) ? v[5] : r;
    r = (i == 6) ? v[6] : r;
    r = (i == 7) ? v[7] : r;
    return r;
}

// One wave32 per tile of 16 unique-ids. Uses V_WMMA_F32_16X16X4_F32 to batch
// the 5 row-wise dot products (||a||^2, ||p||^2, ||n||^2, a.p, a.n) for 16
// rows at a time; diagonals of the 16x16 products are the per-row dots.
//
// 32-bit A 16x4 layout:  lane l, VGPR j holds A[l&15][2*(l>>4)+j]
// 32-bit B 4x16 layout:  lane l, VGPR j holds B[2*(l>>4)+j][l&15]
// => with B = (rows indexed by u)^T, both operands are fed by the SAME
//    per-lane float2 load: row (u_base + (l&15)), cols k0 + 2*(l>>4) .. +1.
__global__ __launch_bounds__(32) void reid_tile_kernel(
    const float* __restrict__ emb,
    const int*   __restrict__ aidx,
    const int*   __restrict__ pidx,
    const int*   __restrict__ nidx,
    const float* __restrict__ has_pos,
    const float* __restrict__ has_neg,
    float*       __restrict__ ws,
    int U, int D)
{
    const int lane = (int)threadIdx.x;        // 0..31
    const int tile = (int)blockIdx.x;         // 16 u's per tile
    const int m    = lane & 15;               // row within tile
    const int kh   = (lane >> 4) << 1;        // 0 or 2: K sub-offset

    int u  = tile * 16 + m;
    int uc = (u < U) ? u : (U - 1);           // clamp so EXEC stays all-ones

    const float* rowA = emb + (size_t)aidx[uc] * (size_t)D;
    const float* rowP = emb + (size_t)pidx[uc] * (size_t)D;
    const float* rowN = emb + (size_t)nidx[uc] * (size_t)D;

    v8f c_aa = {}; v8f c_pp = {}; v8f c_nn = {}; v8f c_ap = {}; v8f c_an = {};

    for (int k0 = 0; k0 < D; k0 += 4) {
        v2f a = *(const v2f*)(rowA + k0 + kh);
        v2f p = *(const v2f*)(rowP + k0 + kh);
        v2f n = *(const v2f*)(rowN + k0 + kh);
        // 8 args: (neg_a, A, neg_b, B, c_mod, C, reuse_a, reuse_b)
        c_aa = __builtin_amdgcn_wmma_f32_16x16x4_f32(false, a, false, a, (short)0, c_aa, false, false);
        c_pp = __builtin_amdgcn_wmma_f32_16x16x4_f32(false, p, false, p, (short)0, c_pp, false, false);
        c_nn = __builtin_amdgcn_wmma_f32_16x16x4_f32(false, n, false, n, (short)0, c_nn, false, false);
        c_ap = __builtin_amdgcn_wmma_f32_16x16x4_f32(false, a, false, p, (short)0, c_ap, false, false);
        c_an = __builtin_amdgcn_wmma_f32_16x16x4_f32(false, a, false, n, (short)0, c_an, false, false);
    }

    // 16x16 f32 C/D layout: VGPR r holds row M=r (lanes 0-15, N=lane) and
    // row M=8+r (lanes 16-31, N=lane-16). Diagonal D[m][m]:
    //   m < 8  -> vgpr m,   lane m        (lanes 0..7)
    //   m >= 8 -> vgpr m-8, lane 16+m     (lanes 24..31)
    int dn = -1, di = 0;
    if (lane < 8)        { dn = lane;      di = lane;      }
    else if (lane >= 24) { dn = lane - 16; di = lane - 24; }

    if (dn >= 0) {
        int uu = tile * 16 + dn;
        if (uu < U) {
            float naa = vsel8(c_aa, di);
            float npp = vsel8(c_pp, di);
            float nnn = vsel8(c_nn, di);
            float dap = vsel8(c_ap, di);
            float dan = vsel8(c_an, di);
            float ia  = 1.0f / fmaxf(sqrtf(naa), REID_EPS);
            float ip  = 1.0f / fmaxf(sqrtf(npp), REID_EPS);
            float inn = 1.0f / fmaxf(sqrtf(nnn), REID_EPS);
            float cap = dap * ia * ip;     // cosine(anchor, pos)
            float can = dan * ia * inn;    // cosine(anchor, neg)
            float pd  = fmaxf(1.0f - cap, 0.0f);
            float nd  = fmaxf(2.0f - (1.0f - can), 0.0f);
            ws[uu] = pd * has_pos[uu] + nd * has_neg[uu];
        }
    }
}

// Deterministic single-block reduction of U per-u contributions.
__global__ __launch_bounds__(256) void reid_reduce_kernel(
    const float* __restrict__ ws, float* __restrict__ out, int U)
{
    __shared__ float s[256];
    float acc = 0.0f;
    for (int i = (int)threadIdx.x; i < U; i += 256) acc += ws[i];
    s[threadIdx.x] = acc;
    __syncthreads();
    for (int off = 128; off > 0; off >>= 1) {
        if ((int)threadIdx.x < off) s[threadIdx.x] += s[threadIdx.x + off];
        __syncthreads();
    }
    if (threadIdx.x == 0) out[0] = s[0] / (float)U;
}

extern "C" void kernel_launch(void* const* d_in, const int* in_sizes, int n_in,
                              void* d_out, int out_size, void* d_ws, size_t ws_size,
                              hipStream_t stream) {
    // Inputs (setup_inputs order):
    // 0: ids_x (int64, N)       -- unused by the loss
    // 1: embeddings (f32, N*D)
    // 2: anchor_idx (i32, U)
    // 3: pos_idx (i32, U)
    // 4: neg_idx (i32, U)
    // 5: has_pos (f32, U)
    // 6: has_neg (f32, U)
    // 7: T, 8: P               -- unused
    const float* emb  = (const float*)d_in[1];
    const int*   aidx = (const int*)d_in[2];
    const int*   pidx = (const int*)d_in[3];
    const int*   nidx = (const int*)d_in[4];
    const float* hp   = (const float*)d_in[5];
    const float* hn   = (const float*)d_in[6];

    const int N = in_sizes[0];
    const int D = in_sizes[1] / N;   // 512
    const int U = in_sizes[2];       // ~2048

    float* ws = (float*)d_ws;        // U floats of per-u contributions

    const int tiles = (U + 15) / 16;
    reid_tile_kernel<<<tiles, 32, 0, stream>>>(emb, aidx, pidx, nidx, hp, hn, ws, U, D);
    reid_reduce_kernel<<<1, 256, 0, stream>>>(ws, (float*)d_out, U);
}